// InteractionPredictionHead_41429254537606
// MI455X (gfx1250) — compile-verified
//
#include <hip/hip_runtime.h>

typedef __attribute__((ext_vector_type(2))) float v2f;
typedef __attribute__((ext_vector_type(8))) float v8f;

#define TILE_E 32
#define HIDN 256
#define FSTRIDE 524   // 515 padded; multiple of 4 (aligned b128), 524%64=12 -> conflict-free
#define H1S 260
#define H2S 132
#define H3S 68

__device__ __forceinline__ v8f wmma4(v2f a, v2f b, v8f c) {
  // V_WMMA_F32_16X16X4_F32: full-fp32 matrix op, D = A(16x4) x B(4x16) + C
  return __builtin_amdgcn_wmma_f32_16x16x4_f32(false, a, false, b, (short)0, c,
                                               false, false);
}

// 2M x 2N register-blocked GEMM: A (LDS) reused across both N tiles,
// B (global, L2-hot) reused across both M tiles -> 1.5 mem instr / WMMA.
__device__ __forceinline__ void gemm_2x2(const float* __restrict__ A, int ldA,
                                         const float* __restrict__ B, int ldB,
                                         int m0a, int m0b, int n0a, int n0b,
                                         int K, int lane,
                                         v8f& c00, v8f& c01, v8f& c10, v8f& c11) {
  const int hi = lane >> 4;
  const int lo = lane & 15;
  const float* a0 = A + (m0a + lo) * ldA + 2 * hi;
  const float* a1 = A + (m0b + lo) * ldA + 2 * hi;
  const float* b0 = B + (size_t)(2 * hi) * ldB + n0a + lo;
  const float* b1 = B + (size_t)(2 * hi) * ldB + n0b + lo;
#pragma unroll 4
  for (int k = 0; k < K; k += 4) {
    v2f av0; av0.x = a0[k]; av0.y = a0[k + 1];
    v2f av1; av1.x = a1[k]; av1.y = a1[k + 1];
    v2f bv0; bv0.x = b0[(size_t)k * ldB]; bv0.y = b0[(size_t)k * ldB + ldB];
    v2f bv1; bv1.x = b1[(size_t)k * ldB]; bv1.y = b1[(size_t)k * ldB + ldB];
    c00 = wmma4(av0, bv0, c00);
    c01 = wmma4(av0, bv1, c01);
    c10 = wmma4(av1, bv0, c10);
    c11 = wmma4(av1, bv1, c11);
  }
}

// 2M x 1N: B reused across both M tiles.
__device__ __forceinline__ void gemm_2x1(const float* __restrict__ A, int ldA,
                                         const float* __restrict__ B, int ldB,
                                         int m0a, int m0b, int n0, int K, int lane,
                                         v8f& c0, v8f& c1) {
  const int hi = lane >> 4;
  const int lo = lane & 15;
  const float* a0 = A + (m0a + lo) * ldA + 2 * hi;
  const float* a1 = A + (m0b + lo) * ldA + 2 * hi;
  const float* b  = B + (size_t)(2 * hi) * ldB + n0 + lo;
#pragma unroll 4
  for (int k = 0; k < K; k += 4) {
    v2f av0; av0.x = a0[k]; av0.y = a0[k + 1];
    v2f av1; av1.x = a1[k]; av1.y = a1[k + 1];
    v2f bv;  bv.x = b[(size_t)k * ldB]; bv.y = b[(size_t)k * ldB + ldB];
    c0 = wmma4(av0, bv, c0);
    c1 = wmma4(av1, bv, c1);
  }
}

__device__ __forceinline__ v8f gemm_1x1(const float* __restrict__ A, int ldA,
                                        const float* __restrict__ B, int ldB,
                                        int m0, int n0, int K, int lane, v8f acc) {
  const int hi = lane >> 4;
  const int lo = lane & 15;
  const float* a = A + (m0 + lo) * ldA + 2 * hi;
  const float* b = B + (size_t)(2 * hi) * ldB + n0 + lo;
#pragma unroll 4
  for (int k = 0; k < K; k += 4) {
    v2f av; av.x = a[k]; av.y = a[k + 1];
    v2f bv; bv.x = b[(size_t)k * ldB]; bv.y = b[(size_t)k * ldB + ldB];
    acc = wmma4(av, bv, acc);
  }
  return acc;
}

// C/D layout store: vgpr v -> row m0 + v + 8*hi, col n0 + lo
__device__ __forceinline__ void store_bias(float* __restrict__ C, int ldC,
                                           int m0, int n0, int lane, v8f acc,
                                           float bb, bool relu) {
  const int hi = lane >> 4;
  const int lo = lane & 15;
#pragma unroll
  for (int v = 0; v < 8; ++v) {
    float x = acc[v] + bb;
    if (relu) x = fmaxf(x, 0.0f);
    C[(m0 + v + 8 * hi) * ldC + n0 + lo] = x;
  }
}

__global__ __launch_bounds__(256) void edge_mlp_kernel(
    const float* __restrict__ nodes,      // [N_NODES][256]
    const int* __restrict__ edge_index,   // [2][E]
    const float* __restrict__ attr,       // [E][3]
    const float* __restrict__ W1, const float* __restrict__ b1,  // [515][256],[256]
    const float* __restrict__ W2, const float* __restrict__ b2,  // [256][128],[128]
    const float* __restrict__ W3, const float* __restrict__ b3,  // [128][64],[64]
    const float* __restrict__ W4, const float* __restrict__ b4,  // [64][3],[3]
    float* __restrict__ out, int E) {
  extern __shared__ float smem[];
  float* feats  = smem;                               // [32][524]
  float* h1     = smem + TILE_E * FSTRIDE;            // [32][260]
  float* w1tail = h1 + TILE_E * H1S;                  // [4][256] rows 512..514 + zero
  float* h2     = smem;                               // overlay feats: [32][132]
  float* h3     = smem + TILE_E * H2S;                // overlay feats: [32][68]

  __shared__ int sSrc[TILE_E], sDst[TILE_E];

  const int tid  = threadIdx.x;
  const int lane = tid & 31;
  const int wave = tid >> 5;
  const long long base = (long long)blockIdx.x * TILE_E;

  if (tid < TILE_E) {
    long long e = base + tid;
    int s = 0, d = 0;
    if (e < E) {
      s = edge_index[e];
      d = edge_index[(long long)E + e];
    }
    sSrc[tid] = s;
    sDst[tid] = d;
  }
  __syncthreads();

  // ---- gather embeddings as float4: 32 edges x 128 vec4 (64 src + 64 dst)
  for (int v = tid; v < TILE_E * 128; v += 256) {
    int er = v >> 7;
    int c4 = v & 127;
    long long e = base + er;
    float4 val = make_float4(0.f, 0.f, 0.f, 0.f);
    int col;
    if (c4 < 64) {
      col = c4 * 4;
      if (e < E) val = *(const float4*)(nodes + (long long)sSrc[er] * HIDN + col);
    } else {
      col = (c4 - 64) * 4 + HIDN;
      if (e < E) val = *(const float4*)(nodes + (long long)sDst[er] * HIDN + (col - HIDN));
    }
    *(float4*)(feats + er * FSTRIDE + col) = val;
  }
  // ---- ragged tail: cols 512..523 = attr (3) + zero pad (9)
  if (tid < TILE_E * 12) {
    int er = tid / 12;
    int c  = 512 + (tid - er * 12);
    long long e = base + er;
    float v = 0.0f;
    if (e < E && c < 515) v = attr[e * 3 + (c - 512)];
    feats[er * FSTRIDE + c] = v;
  }
  // ---- stage W1 tail rows (512..514) + one zero row so K-tail is a clean x4 step
  for (int i = tid; i < 4 * HIDN; i += 256) {
    int r = i >> 8, c = i & (HIDN - 1);
    w1tail[i] = (r < 3) ? W1[(size_t)(512 + r) * HIDN + c] : 0.0f;
  }
  __syncthreads();

  const int hi = lane >> 4;
  const int lo = lane & 15;

  // ---- layer 1: [32,515]x[515,256] -> relu -> h1
  // 32 tiles = 8 waves x (2 mt x 2 nt) register-blocked superjobs
  {
    int n0a = wave * 32;
    int n0b = n0a + 16;
    v8f c00 = {}, c01 = {}, c10 = {}, c11 = {};
    gemm_2x2(feats, FSTRIDE, W1, HIDN, 0, 16, n0a, n0b, 512, lane, c00, c01, c10, c11);
    gemm_2x2(feats + 512, FSTRIDE, w1tail, HIDN, 0, 16, n0a, n0b, 4, lane, c00, c01, c10, c11);
    float ba = b1[n0a + lo];
    float bbv = b1[n0b + lo];
    store_bias(h1, H1S, 0,  n0a, lane, c00, ba,  true);
    store_bias(h1, H1S, 0,  n0b, lane, c01, bbv, true);
    store_bias(h1, H1S, 16, n0a, lane, c10, ba,  true);
    store_bias(h1, H1S, 16, n0b, lane, c11, bbv, true);
  }
  __syncthreads();

  // ---- layer 2: [32,256]x[256,128] (+bias, no relu) -> h2
  // 16 tiles = 8 waves x (2 mt x 1 nt)
  {
    int n0 = wave * 16;
    v8f c0 = {}, c1 = {};
    gemm_2x1(h1, H1S, W2, 128, 0, 16, n0, HIDN, lane, c0, c1);
    float bbv = b2[n0 + lo];
    store_bias(h2, H2S, 0,  n0, lane, c0, bbv, false);
    store_bias(h2, H2S, 16, n0, lane, c1, bbv, false);
  }
  __syncthreads();

  // ---- layer 3: [32,128]x[128,64] -> relu -> h3 ; 8 tiles, 1/wave
  {
    int mt = wave >> 2, nt = wave & 3;
    v8f acc = {};
    acc = gemm_1x1(h2, H2S, W3, 64, mt * 16, nt * 16, 128, lane, acc);
    float bbv = b3[nt * 16 + lo];
    store_bias(h3, H3S, mt * 16, nt * 16, lane, acc, bbv, true);
  }
  __syncthreads();

  // ---- layer 4: [32,64]x[64,3] + b4 -> logits (too skinny for WMMA)
  if (tid < TILE_E * 3) {
    int er = tid / 3, o = tid - er * 3;
    long long e = base + er;
    if (e < E) {
      float s = b4[o];
#pragma unroll 8
      for (int k = 0; k < 64; ++k) s += h3[er * H3S + k] * W4[k * 3 + o];
      out[e * 3 + o] = s;
    }
  }
}

extern "C" void kernel_launch(void* const* d_in, const int* in_sizes, int n_in,
                              void* d_out, int out_size, void* d_ws, size_t ws_size,
                              hipStream_t stream) {
  (void)n_in; (void)d_ws; (void)ws_size; (void)out_size;
  const float* nodes      = (const float*)d_in[0];
  const int*   edge_index = (const int*)d_in[1];
  const float* attr       = (const float*)d_in[2];
  const float* W1 = (const float*)d_in[3];
  const float* b1 = (const float*)d_in[4];
  const float* W2 = (const float*)d_in[5];
  const float* b2 = (const float*)d_in[6];
  const float* W3 = (const float*)d_in[7];
  const float* b3 = (const float*)d_in[8];
  const float* W4 = (const float*)d_in[9];
  const float* b4 = (const float*)d_in[10];
  float* out = (float*)d_out;

  const int E = in_sizes[1] / 2;  // edge_index is [2][E]
  const int grid = (E + TILE_E - 1) / TILE_E;  // 9375 for E=300000
  const size_t shbytes =
      (size_t)(TILE_E * FSTRIDE + TILE_E * H1S + 4 * HIDN) * sizeof(float);

  // CDNA5 allows up to 320KB LDS per workgroup; opt in past the 64KB default.
  (void)hipFuncSetAttribute((const void*)edge_mlp_kernel,
                            hipFuncAttributeMaxDynamicSharedMemorySize,
                            (int)shbytes);

  edge_mlp_kernel<<<grid, 256, shbytes, stream>>>(
      nodes, edge_index, attr, W1, b1, W2, b2, W3, b3, W4, b4, out, E);
}